// ResNet50ROIHead_12489764897441
// MI455X (gfx1250) — compile-verified
//
#include <hip/hip_runtime.h>
#include <hip/hip_bf16.h>

typedef __attribute__((ext_vector_type(2))) float v2f;
typedef __attribute__((ext_vector_type(8))) float v8f;
typedef __attribute__((ext_vector_type(4))) unsigned int u32x4;
typedef __attribute__((ext_vector_type(8))) int i32x8;
typedef __attribute__((ext_vector_type(4))) int i32x4;

#define C_IN  1024
#define HFM   38
#define WFM   38
#define PP    14
#define HW    196          // 14*14 pooled positions per ROI
#define MPAD  208          // HW padded to 13 WMMA row-tiles
#define MT    13           // M tiles of 16
#define NBLK  128          // N columns per workgroup (8 waves x 16)
#define DOUT  2048
#define KB    16           // K slab staged in LDS (double buffered)
#define KPB   18           // LDS row stride (16 + 2 TDM pad dwords)
#define NROI  128

#if __has_builtin(__builtin_amdgcn_tensor_load_to_lds)
#define HAVE_TDM 1
#else
#define HAVE_TDM 0
#endif

// ---------------------------------------------------------------------------
// Kernel 1: ROI max pool (torchvision RoIPool semantics). Output layout is
// M-major / channel-contiguous: pool[r][m][c], m in [0,208), rows 196..207
// zeroed (GEMM padding). Channel-contiguous rows make the GEMM A slab the
// same TDM descriptor shape as the weight slab (16 contiguous K elements).
// ---------------------------------------------------------------------------
__global__ void roi_pool_kernel(const float* __restrict__ x,
                                const float* __restrict__ rois,
                                const int*  __restrict__ roi_idx,
                                const int*  __restrict__ img_h,
                                const int*  __restrict__ img_w,
                                float* __restrict__ pool) {
  const int r  = blockIdx.x;          // 128 ROIs
  const int cg = blockIdx.y;          // 16 groups of 64 channels
  const int b  = roi_idx[r];
  const float sx = (float)WFM / (float)img_w[0];
  const float sy = (float)HFM / (float)img_h[0];
  const float x1 = rois[r * 4 + 0] * sx, y1 = rois[r * 4 + 1] * sy;
  const float x2 = rois[r * 4 + 2] * sx, y2 = rois[r * 4 + 3] * sy;
  // jnp.round == round-half-even == rintf (RNE default)
  const int sw = (int)rintf(x1), sh = (int)rintf(y1);
  const int ew = (int)rintf(x2), eh = (int)rintf(y2);
  const float bw = fmaxf((float)(ew - sw + 1), 1.0f) / (float)PP;
  const float bh = fmaxf((float)(eh - sh + 1), 1.0f) / (float)PP;

  for (int e = threadIdx.x; e < 64 * MPAD; e += blockDim.x) {
    const int cl = e & 63;            // c fastest -> coalesced 256B stores
    const int m  = e >> 6;
    const int c  = cg * 64 + cl;
    float val = 0.0f;
    if (m < HW) {
      const int ph = m / PP, pw = m % PP;
      int hs = (int)floorf((float)ph * bh) + sh;
      int he = (int)ceilf((float)(ph + 1) * bh) + sh;
      int wsr = (int)floorf((float)pw * bw) + sw;
      int wer = (int)ceilf((float)(pw + 1) * bw) + sw;
      hs  = min(max(hs, 0), HFM);  he  = min(max(he, 0), HFM);
      wsr = min(max(wsr, 0), WFM); wer = min(max(wer, 0), WFM);
      if (he > hs && wer > wsr) {
        const float* src = x + (size_t)(b * C_IN + c) * (HFM * WFM);
        float mx = -1e30f;
        for (int h = hs; h < he; ++h)
          for (int w = wsr; w < wer; ++w)
            mx = fmaxf(mx, src[h * WFM + w]);
        val = mx;
      }
    }
    pool[((size_t)r * MPAD + m) * C_IN + c] = val;
  }
}

// ---------------------------------------------------------------------------
// TDM helper: issue a 2-D tensor_load_to_lds per the D# layout
// (CDNA5 ISA). data_size=4B, LDS pad +2dw every 16dw.
// group0: [1:0]=count=1, [63:32]=lds_addr, [120:64]=global_addr, [127:126]=2
// group1: [17:16]=data_size, [20]=pad_en, [24:22]=pad_interval,
//         [31:25]=pad_amount, [79:48]=tensor_dim0, [111:80]=tensor_dim1,
//         [127:112]=tile_dim0, [143:128]=tile_dim1, [207:160]=dim0_stride
// Groups 2/3 (dims 2-4) zero: pure 2-D tile.  6-arg builtin on this
// toolchain: (u32x4, i32x8, i32x4, i32x4, i32x8, i32 cpol).
// ---------------------------------------------------------------------------
#if HAVE_TDM
__device__ __forceinline__ void tdm_load_2d(unsigned int lds_addr,
                                            const void* gptr,
                                            unsigned int flags0,   // data_size/pad bits
                                            unsigned int tdim0, unsigned int tdim1,
                                            unsigned int tile0, unsigned int tile1,
                                            unsigned long long stride0) {
  union { unsigned long long q[2]; u32x4 v; } g0;
  g0.q[0] = 1ull | ((unsigned long long)lds_addr << 32);
  g0.q[1] = ((unsigned long long)(size_t)gptr & 0x01FFFFFFFFFFFFFFull) | (2ull << 62);
  union { unsigned long long q[4]; i32x8 v; } g1;
  g1.q[0] = (unsigned long long)flags0 | ((unsigned long long)(tdim0 & 0xFFFFu) << 48);
  g1.q[1] = ((unsigned long long)tdim0 >> 16) | ((unsigned long long)tdim1 << 16) |
            ((unsigned long long)(tile0 & 0xFFFFu) << 48);
  g1.q[2] = (unsigned long long)(tile1 & 0xFFFFu) |          // tile_dim2 = 0
            ((stride0 & 0xFFFFFFFFull) << 32);
  g1.q[3] = (stride0 >> 32) & 0xFFFFull;                     // dim1_stride = 0 (2D)
  const i32x4 z4 = {0, 0, 0, 0};
  const i32x8 z8 = {0, 0, 0, 0, 0, 0, 0, 0};
  __builtin_amdgcn_tensor_load_to_lds(g0.v, g1.v, z4, z4, z8, 0);
}
#endif

// shared compute: one K-slab of WMMAs on staged LDS buffers.
// Both slabs are [row][KPB] with K contiguous: every fragment is one
// 8B-aligned ds_load_b64, bank-conflict-free (18*l mod 64 distinct).
__device__ __forceinline__ void compute_slab(const float* lA, const float* lB,
                                             v8f* acc, int row16, int hi, int nb) {
#pragma unroll
  for (int ks = 0; ks < KB; ks += 4) {
    // A 16x4 fp32 frag: lanes 0-15 -> K = ks,ks+1 ; lanes 16-31 -> K+2
    const int k0 = ks + 2 * hi;
    const v2f bfrag = *(const v2f*)&lB[(nb + row16) * KPB + k0];
#pragma unroll
    for (int mt = 0; mt < MT; ++mt) {
      const v2f afrag = *(const v2f*)&lA[(mt * 16 + row16) * KPB + k0];
      acc[mt] = __builtin_amdgcn_wmma_f32_16x16x4_f32(
          false, afrag, false, bfrag, (short)0, acc[mt], false, false);
    }
  }
}

// ---------------------------------------------------------------------------
// Kernel 2: fp32 WMMA GEMM (per-ROI 208x128 tile over K=1024) fused with
// bias + ReLU + spatial mean -> fc7[128][2048].
// TDM double-buffers both LDS slabs: wave 0 issues tensor_load_to_lds for
// slab i+1 while all 8 waves run WMMAs on slab i; s_wait_tensorcnt + barrier
// hands the buffer over.
// ---------------------------------------------------------------------------
__global__ __launch_bounds__(256) void head_gemm_kernel(
    const float* __restrict__ pool,   // [128][208][1024]
    const float* __restrict__ Wc,     // [2048][1024]
    const float* __restrict__ bc,     // [2048]
    float* __restrict__ fc7) {        // [128][2048]
  __shared__ float lsA[2][MPAD][KPB]; // 2 x 14976 B  (TDM pad -> 18 dw rows)
  __shared__ float lsB[2][NBLK][KPB]; // 2 x  9216 B

  const int r     = blockIdx.x;
  const int dbase = blockIdx.y * NBLK;
  const int t     = threadIdx.x;
  const int wv    = t >> 5;           // wave id (wave32)
  const int lane  = t & 31;
  const int row16 = lane & 15;
  const int hi    = lane >> 4;
  const int nb    = wv * 16;

  v8f acc[MT];
  const v8f vzero = {0.f, 0.f, 0.f, 0.f, 0.f, 0.f, 0.f, 0.f};
#pragma unroll
  for (int i = 0; i < MT; ++i) acc[i] = vzero;

  const float* poolR = pool + (size_t)r * MPAD * C_IN;
  const float* WcR   = Wc + (size_t)dbase * C_IN;

#if HAVE_TDM
  const unsigned int TD_FLAGS = (2u << 16)        // data_size = 4B
                              | (1u << 20)        // pad_enable
                              | (3u << 22)        // pad_interval: every 16 dw
                              | (1u << 25);       // pad_amount: +2 dw
  const unsigned int ldsA[2] = {(unsigned int)(size_t)&lsA[0][0][0],
                                (unsigned int)(size_t)&lsA[1][0][0]};
  const unsigned int ldsB[2] = {(unsigned int)(size_t)&lsB[0][0][0],
                                (unsigned int)(size_t)&lsB[1][0][0]};
  if (wv == 0) {
    tdm_load_2d(ldsA[0], poolR, TD_FLAGS, KB, MPAD, KB, MPAD, C_IN);
    tdm_load_2d(ldsB[0], WcR,   TD_FLAGS, KB, NBLK, KB, NBLK, C_IN);
    __builtin_amdgcn_s_wait_tensorcnt(0);
  }
  __syncthreads();
  int cur = 0;
  for (int kb = 0; kb < C_IN; kb += KB) {
    if (wv == 0 && (kb + KB) < C_IN) {    // prefetch next slab into other buffer
      tdm_load_2d(ldsA[cur ^ 1], poolR + (kb + KB),
                  TD_FLAGS, KB, MPAD, KB, MPAD, C_IN);
      tdm_load_2d(ldsB[cur ^ 1], WcR + (kb + KB),
                  TD_FLAGS, KB, NBLK, KB, NBLK, C_IN);
    }
    compute_slab(&lsA[cur][0][0], &lsB[cur][0][0], acc, row16, hi, nb);
    if (wv == 0) __builtin_amdgcn_s_wait_tensorcnt(0);
    __syncthreads();
    cur ^= 1;
  }
#else
  for (int kb = 0; kb < C_IN; kb += KB) {
#pragma unroll
    for (int i = 0; i < (KB * MPAD) / 256; ++i) {     // 13 iters
      const int e = t + i * 256;
      const int k = e & (KB - 1), m = e >> 4;
      lsA[0][m][k] = poolR[(size_t)m * C_IN + kb + k];
    }
#pragma unroll
    for (int i = 0; i < (KB * NBLK) / 256; ++i) {     // 8 iters
      const int e = t + i * 256;
      const int k = e & (KB - 1), n = e >> 4;
      lsB[0][n][k] = WcR[(size_t)n * C_IN + kb + k];
    }
    __syncthreads();
    compute_slab(&lsA[0][0][0], &lsB[0][0][0], acc, row16, hi, nb);
    __syncthreads();
  }
#endif

  // Epilogue: fc7[r][d] = mean_{m<196} relu(acc[m][d] + bc[d])
  // C/D layout: lane l, vgpr v -> (M = v + 8*(l>>4), N = l&15)
  const float bias = bc[dbase + nb + row16];
  float colsum = 0.0f;
#pragma unroll
  for (int mt = 0; mt < MT; ++mt) {
#pragma unroll
    for (int v = 0; v < 8; ++v) {
      const int Mg = mt * 16 + hi * 8 + v;
      if (Mg < HW) colsum += fmaxf(acc[mt][v] + bias, 0.0f);
    }
  }
  colsum += __shfl_xor(colsum, 16);   // combine the two half-wave row groups
  if (hi == 0)
    fc7[(size_t)r * DOUT + dbase + nb + row16] = colsum * (1.0f / 196.0f);
}

// ---------------------------------------------------------------------------
// Kernel 3: output linears. out = [locs 128x84 | scores 128x21] flat.
// ---------------------------------------------------------------------------
__global__ void out_linear_kernel(const float* __restrict__ fc7,
                                  const float* __restrict__ W_loc,
                                  const float* __restrict__ b_loc,
                                  const float* __restrict__ W_score,
                                  const float* __restrict__ b_score,
                                  float* __restrict__ out) {
  __shared__ float f[DOUT];
  const int r = blockIdx.x;
  for (int i = threadIdx.x; i < DOUT; i += blockDim.x)
    f[i] = fc7[(size_t)r * DOUT + i];
  __syncthreads();
  const int j = threadIdx.x;
  if (j >= 105) return;
  const float* wrow;
  float bias;
  float* dst;
  if (j < 84) {
    wrow = W_loc + (size_t)j * DOUT;
    bias = b_loc[j];
    dst  = out + (size_t)r * 84 + j;
  } else {
    const int j2 = j - 84;
    wrow = W_score + (size_t)j2 * DOUT;
    bias = b_score[j2];
    dst  = out + (size_t)NROI * 84 + (size_t)r * 21 + j2;
  }
  float a0 = 0.f, a1 = 0.f, a2 = 0.f, a3 = 0.f;
  for (int k = 0; k < DOUT; k += 4) {
    a0 += f[k + 0] * wrow[k + 0];
    a1 += f[k + 1] * wrow[k + 1];
    a2 += f[k + 2] * wrow[k + 2];
    a3 += f[k + 3] * wrow[k + 3];
  }
  *dst = (a0 + a1) + (a2 + a3) + bias;
}

// ---------------------------------------------------------------------------
extern "C" void kernel_launch(void* const* d_in, const int* in_sizes, int n_in,
                              void* d_out, int out_size, void* d_ws, size_t ws_size,
                              hipStream_t stream) {
  const float* x       = (const float*)d_in[0];   // [2,1024,38,38]
  const float* rois    = (const float*)d_in[1];   // [2,64,4]
  const int*   roi_idx = (const int*)  d_in[2];   // [2,64]
  const float* Wc      = (const float*)d_in[3];   // [2048,1024]
  const float* bc      = (const float*)d_in[4];   // [2048]
  const float* W_loc   = (const float*)d_in[5];   // [84,2048]
  const float* b_loc   = (const float*)d_in[6];   // [84]
  const float* W_score = (const float*)d_in[7];   // [21,2048]
  const float* b_score = (const float*)d_in[8];   // [21]
  const int*   img_h   = (const int*)  d_in[9];   // scalar
  const int*   img_w   = (const int*)  d_in[10];  // scalar
  float* out = (float*)d_out;

  float* pool = (float*)d_ws;                          // 128*208*1024 f32
  float* fc7  = pool + (size_t)NROI * MPAD * C_IN;     // 128*2048 f32

  roi_pool_kernel<<<dim3(NROI, C_IN / 64), 256, 0, stream>>>(
      x, rois, roi_idx, img_h, img_w, pool);

  head_gemm_kernel<<<dim3(NROI, DOUT / NBLK), 256, 0, stream>>>(
      pool, Wc, bc, fc7);

  out_linear_kernel<<<dim3(NROI), 128, 0, stream>>>(
      fc7, W_loc, b_loc, W_score, b_score, out);
}